// LSM_65979287601801
// MI455X (gfx1250) — compile-verified
//
#include <hip/hip_runtime.h>
#include <hip/hip_bf16.h>
#include <cmath>

#define K1 15
#define M_PAIRS 8192
#define TOTAL_K (K1 + 2 * M_PAIRS)   // 16399

typedef __attribute__((ext_vector_type(2)))  float    v2f;
typedef __attribute__((ext_vector_type(8)))  float    v8f;

// ---------------------------------------------------------------------------
// Zero the workspace: s[TOTAL_K] bins + 1 accumulator.
// ---------------------------------------------------------------------------
__global__ void zero_ws_kernel(float* __restrict__ buf, int n) {
    int i = blockIdx.x * blockDim.x + threadIdx.x;
    if (i < n) buf[i] = 0.0f;
}

// ---------------------------------------------------------------------------
// Segment sum: s[k_i[i]] += exp(gamma[n_j[i]]).
// LDS-privatized histogram (65.6KB dynamic LDS, fine with 320KB/WGP),
// ds_add_f32 atomics in LDS, sparse flush of non-zero bins to global.
// Index streams vectorized as int4 (global_load_b128) — 4 elements per
// iteration so expf chains, gathers and LDS atomics overlap.
// ---------------------------------------------------------------------------
__global__ void segment_sum_kernel(const float* __restrict__ gamma,
                                   const int*   __restrict__ k_i,
                                   const int*   __restrict__ n_j,
                                   float*       __restrict__ s_out,
                                   int n) {
    extern __shared__ float sh[];   // TOTAL_K floats
    for (int t = threadIdx.x; t < TOTAL_K; t += blockDim.x) sh[t] = 0.0f;
    __syncthreads();

    const int gid    = blockIdx.x * blockDim.x + threadIdx.x;
    const int stride = gridDim.x * blockDim.x;
    const int nv4    = n >> 2;

    const int4* __restrict__ nj4p = (const int4*)n_j;
    const int4* __restrict__ ki4p = (const int4*)k_i;

    for (int i = gid; i < nv4; i += stride) {
        int4 nj = nj4p[i];                 // global_load_b128
        int4 ki = ki4p[i];                 // global_load_b128
        float g0 = gamma[nj.x];
        float g1 = gamma[nj.y];
        float g2 = gamma[nj.z];
        float g3 = gamma[nj.w];
        atomicAdd(&sh[ki.x], expf(g0));    // ds_add_f32
        atomicAdd(&sh[ki.y], expf(g1));
        atomicAdd(&sh[ki.z], expf(g2));
        atomicAdd(&sh[ki.w], expf(g3));
    }
    // tail (n not divisible by 4)
    for (int i = (nv4 << 2) + gid; i < n; i += stride) {
        atomicAdd(&sh[k_i[i]], expf(gamma[n_j[i]]));
    }
    __syncthreads();

    for (int t = threadIdx.x; t < TOTAL_K; t += blockDim.x) {
        float v = sh[t];
        if (v != 0.0f) atomicAdd(&s_out[t], v);
    }
}

// ---------------------------------------------------------------------------
// Layer-1 pairwise term via WMMA Gram matrix.
// One wave (32 threads). G = C * C^T for the zero-padded 16x8 centroid
// matrix, computed with two chained V_WMMA_F32_16X16X4_F32 (K=0..3, K=4..7).
// dist^2(i,j) = nrm[i] + nrm[j] - 2*G[i][j].
// C/D layout: lane<16 -> col j=lane, vgpr r -> row r; lanes 16-31 -> rows 8..15.
// ---------------------------------------------------------------------------
__global__ void layer1_wmma_kernel(const float* __restrict__ c1,
                                   const float* __restrict__ s,
                                   float*       __restrict__ accum) {
    __shared__ float nrm[16];
    __shared__ float sv[16];

    int lane = threadIdx.x;       // 0..31
    int m    = lane & 15;

    if (lane < 16) {
        float nv = 0.0f, svv = 0.0f;
        if (lane < K1) {
            #pragma unroll
            for (int k = 0; k < 8; ++k) { float c = c1[lane * 8 + k]; nv += c * c; }
            svv = s[lane];
        }
        nrm[lane] = nv;
        sv[lane]  = svv;
    }
    __syncthreads();

    v8f d = {};

    // A 16x4 f32 layout -> lane<16: (K0,K1) of row m; lane>=16: (K2,K3).
    // For a symmetric Gram, B (4x16 = C^T) has the identical register image.
    int koff = (lane >> 4) << 1;          // 0 or 2
    v2f a0 = {0.0f, 0.0f};
    v2f a1 = {0.0f, 0.0f};
    if (m < K1) {
        a0[0] = c1[m * 8 + koff];
        a0[1] = c1[m * 8 + koff + 1];
        a1[0] = c1[m * 8 + 4 + koff];
        a1[1] = c1[m * 8 + 4 + koff + 1];
    }
    d = __builtin_amdgcn_wmma_f32_16x16x4_f32(false, a0, false, a0,
                                              (short)0, d, false, false);
    d = __builtin_amdgcn_wmma_f32_16x16x4_f32(false, a1, false, a1,
                                              (short)0, d, false, false);

    // Upper-triangular accumulation: this lane owns column j = m,
    // rows i = ib..ib+7. Pairs need i < j and j <= 14 (row 15 is padding).
    float acc = 0.0f;
    int   j   = m;
    int   ib  = (lane >> 4) * 8;
    if (j >= 1 && j < K1) {
        #pragma unroll
        for (int r = 0; r < 8; ++r) {
            int i = ib + r;
            if (i < j) {
                float d2  = nrm[i] + nrm[j] - 2.0f * d[r];
                float val = expf(-sqrtf(fmaxf(d2, 0.0f)));
                acc += val * sv[i] * sv[j];
            }
        }
    }

    // wave32 reduction
    #pragma unroll
    for (int off = 16; off > 0; off >>= 1)
        acc += __shfl_down(acc, off, 32);
    if (lane == 0) atomicAdd(accum, acc);
}

// ---------------------------------------------------------------------------
// Sub-centroid pairs: 8192 pairs, D=8. One pair per thread (four contiguous
// float4 loads = global_load_b128 x4), block reduce, single atomic per block.
// ---------------------------------------------------------------------------
__global__ void subpairs_kernel(const float* __restrict__ tc,
                                const float* __restrict__ s,
                                float*       __restrict__ accum) {
    int p = blockIdx.x * blockDim.x + threadIdx.x;
    float term = 0.0f;
    if (p < M_PAIRS) {
        const float4* a4 = (const float4*)(tc + (size_t)p * 16);
        float4 a0 = a4[0], a1 = a4[1];     // row 2p
        float4 b0 = a4[2], b1 = a4[3];     // row 2p+1
        float d2 = 0.0f;
        float df;
        df = a0.x - b0.x; d2 += df * df;
        df = a0.y - b0.y; d2 += df * df;
        df = a0.z - b0.z; d2 += df * df;
        df = a0.w - b0.w; d2 += df * df;
        df = a1.x - b1.x; d2 += df * df;
        df = a1.y - b1.y; d2 += df * df;
        df = a1.z - b1.z; d2 += df * df;
        df = a1.w - b1.w; d2 += df * df;
        float val = expf(-sqrtf(d2));
        term = val * s[K1 + 2 * p] * s[K1 + 2 * p + 1];
    }

    __shared__ float red[256];
    red[threadIdx.x] = term;
    __syncthreads();
    for (int off = blockDim.x >> 1; off > 0; off >>= 1) {
        if ((int)threadIdx.x < off) red[threadIdx.x] += red[threadIdx.x + off];
        __syncthreads();
    }
    if (threadIdx.x == 0) atomicAdd(accum, red[0]);
}

// ---------------------------------------------------------------------------
// theta = exp(bias) * accum
// ---------------------------------------------------------------------------
__global__ void finalize_kernel(const float* __restrict__ bias,
                                const float* __restrict__ accum,
                                float*       __restrict__ out) {
    if (threadIdx.x == 0 && blockIdx.x == 0)
        out[0] = expf(bias[0]) * accum[0];
}

extern "C" void kernel_launch(void* const* d_in, const int* in_sizes, int n_in,
                              void* d_out, int out_size, void* d_ws, size_t ws_size,
                              hipStream_t stream) {
    const float* c1    = (const float*)d_in[0];   // centroids_layer1 [15,8]
    const float* tc    = (const float*)d_in[1];   // total_centroids  [16384,8]
    const float* gamma = (const float*)d_in[2];   // [N]
    const float* bias  = (const float*)d_in[3];   // [1]
    const int*   k_i   = (const int*)d_in[4];     // [N]
    const int*   n_j   = (const int*)d_in[5];     // [N]
    float*       out   = (float*)d_out;

    int n = in_sizes[2];                          // N (gamma length)

    float* s_buf = (float*)d_ws;                  // TOTAL_K floats
    float* accum = s_buf + TOTAL_K;               // 1 float

    int zn = TOTAL_K + 1;
    zero_ws_kernel<<<(zn + 255) / 256, 256, 0, stream>>>(s_buf, zn);

    const int    segBlocks = 128;
    const size_t shbytes   = (size_t)TOTAL_K * sizeof(float);   // 65596 B
    segment_sum_kernel<<<segBlocks, 256, shbytes, stream>>>(gamma, k_i, n_j,
                                                            s_buf, n);

    layer1_wmma_kernel<<<1, 32, 0, stream>>>(c1, s_buf, accum);
    subpairs_kernel<<<(M_PAIRS + 255) / 256, 256, 0, stream>>>(tc, s_buf, accum);
    finalize_kernel<<<1, 32, 0, stream>>>(bias, accum, out);
}